// GATBasedMolecularGraphNeuralNetwork194New_54872502173939
// MI455X (gfx1250) — compile-verified
//
#include <hip/hip_runtime.h>

#define N_NODES 100000
#define N_EDGES 800000
#define NUM_GRAPHS 4096
#define EDGE_DIM 6
#define HID 128
#define EAUG (N_EDGES + N_NODES)

typedef __attribute__((ext_vector_type(2))) float v2f;
typedef __attribute__((ext_vector_type(8))) float v8f;

__device__ __forceinline__ float lrelu(float x, float s) { return x >= 0.f ? x : s * x; }

// Order-preserving float <-> uint encoding for atomicMax on floats
__device__ __forceinline__ unsigned fenc(float f) {
    unsigned u = __float_as_uint(f);
    return (u & 0x80000000u) ? ~u : (u | 0x80000000u);
}
__device__ __forceinline__ float fdec(unsigned u) {
    return (u & 0x80000000u) ? __uint_as_float(u ^ 0x80000000u) : __uint_as_float(~u);
}

// ---------------- utility fills ----------------
__global__ void k_fill(float* p, float v, int n) {
    int i = blockIdx.x * blockDim.x + threadIdx.x;
    if (i < n) p[i] = v;
}
__global__ void k_fill_u32(unsigned* p, unsigned v, int n) {
    int i = blockIdx.x * blockDim.x + threadIdx.x;
    if (i < n) p[i] = v;
}
__global__ void k_init_bias(float* h, const float* __restrict__ b, int n) {
    int i = blockIdx.x * blockDim.x + threadIdx.x;
    if (i < n) h[i] = b[i & (HID - 1)];
}

// ---------------- precompute: deg, mean edge_attr, graph counts ----------------
__global__ void k_deg_ea(const int* __restrict__ dst, const float* __restrict__ ea,
                         float* deg, float* easum) {
    int e = blockIdx.x * blockDim.x + threadIdx.x;
    if (e >= N_EDGES) return;
    int d = dst[e];
    atomicAdd(deg + d, 1.f);
#pragma unroll
    for (int c = 0; c < EDGE_DIM; c++)
        atomicAdd(easum + (size_t)d * EDGE_DIM + c, ea[(size_t)e * EDGE_DIM + c]);
}
__global__ void k_mea_fin(float* mea, const float* __restrict__ deg) {
    int i = blockIdx.x * blockDim.x + threadIdx.x;
    if (i >= N_NODES * EDGE_DIM) return;
    mea[i] /= fmaxf(deg[i / EDGE_DIM], 1.f);
}
__global__ void k_cnt(const int* __restrict__ batch, float* cnt) {
    int i = blockIdx.x * blockDim.x + threadIdx.x;
    if (i < N_NODES) atomicAdd(cnt + batch[i], 1.f);
}

// ---------------- W staging into per-lane WMMA fragment order ----------------
// Wst[((t*32 + lane)*KS + s)*2 + {0,1}] = { W[4s+2h][t*16+l16], W[4s+2h+1][t*16+l16] }
// where h = lane>>4, l16 = lane&15, KS = K/4. One b128 load then covers 2 k-steps/tile.
__global__ void k_stage_w(const float* __restrict__ W, float* __restrict__ Wst,
                          int K, int NO) {
    int ks = K >> 2;
    int total = (NO >> 4) * 32 * ks;
    int i = blockIdx.x * blockDim.x + threadIdx.x;
    if (i >= total) return;
    int s = i % ks;
    int rem = i / ks;
    int l = rem & 31;
    int t = rem >> 5;
    int h = l >> 4, l16 = l & 15;
    int col = t * 16 + l16;
    int k = 4 * s + 2 * h;
    Wst[(size_t)i * 2 + 0] = W[(size_t)k * NO + col];
    Wst[(size_t)i * 2 + 1] = W[(size_t)(k + 1) * NO + col];
}

// ---------------- generic fp32 WMMA GEMM: out[M, NT*16] = act(A[M,K]) @ W[K, NT*16] + bias ----
// One wave computes a 16-row strip across all NT column tiles with V_WMMA_F32_16X16X4_F32.
// B comes pre-staged (k_stage_w): one global_load_b128 per tile per 2 k-steps.
template <int NT, int AIN, int AOUT>
__global__ void k_gemm_wmma(const float* __restrict__ A, const float* __restrict__ Wst,
                            const float* __restrict__ bias, float* __restrict__ out,
                            int M, int K) {
    const int NO = NT * 16;
    int row0 = blockIdx.x * 16;
    if (row0 >= M) return;
    int lane = threadIdx.x & 31;
    int half = lane >> 4;   // 0: K={k,k+1}, 1: K={k+2,k+3}
    int l16  = lane & 15;
    const int KS = K >> 2;  // always even here (K in {48,64,128})
    const float* arow = A + (size_t)(row0 + l16) * K + 2 * half;
    v8f acc[NT] = {};
    for (int s = 0; s < KS; s += 2) {
        int k = 4 * s;
        v2f a0 = *(const v2f*)(arow + k);
        v2f a1 = *(const v2f*)(arow + k + 4);
        if (AIN) {
            a0.x = lrelu(a0.x, 0.01f); a0.y = lrelu(a0.y, 0.01f);
            a1.x = lrelu(a1.x, 0.01f); a1.y = lrelu(a1.y, 0.01f);
        }
        float4 bb[NT];
#pragma unroll
        for (int t = 0; t < NT; t++)
            bb[t] = *(const float4*)(Wst + ((size_t)(t * 32 + lane) * KS + s) * 2);
#pragma unroll
        for (int t = 0; t < NT; t++) {
            v2f b0; b0.x = bb[t].x; b0.y = bb[t].y;
            acc[t] = __builtin_amdgcn_wmma_f32_16x16x4_f32(
                false, a0, false, b0, (short)0, acc[t], false, false);
        }
#pragma unroll
        for (int t = 0; t < NT; t++) {
            v2f b1; b1.x = bb[t].z; b1.y = bb[t].w;
            acc[t] = __builtin_amdgcn_wmma_f32_16x16x4_f32(
                false, a1, false, b1, (short)0, acc[t], false, false);
        }
    }
#pragma unroll
    for (int t = 0; t < NT; t++) {
        int cc = t * 16 + l16;
        float bv = bias[cc];
#pragma unroll
        for (int v = 0; v < 8; v++) {
            float val = acc[t][v] + bv;
            if (AOUT) val = lrelu(val, 0.01f);
            out[(size_t)(row0 + v + half * 8) * NO + cc] = val;
        }
    }
}

// ---------------- edge phase (wave per augmented edge) ----------------
__global__ void k_edge_score(const int* __restrict__ src, const int* __restrict__ dst,
                             const float* __restrict__ ea, const float* __restrict__ mea,
                             const float* __restrict__ xl, const float* __restrict__ xr,
                             const float* __restrict__ We, const float* __restrict__ att,
                             float* __restrict__ score, unsigned* __restrict__ nmax) {
    int wid  = blockIdx.x * (blockDim.x >> 5) + (threadIdx.x >> 5);
    int lane = threadIdx.x & 31;
    if (wid >= EAUG) return;
    int s, d;
    const float* e6;
    if (wid < N_EDGES) { s = src[wid]; d = dst[wid]; e6 = ea + (size_t)wid * EDGE_DIM; }
    else               { s = wid - N_EDGES; d = s;   e6 = mea + (size_t)s * EDGE_DIM; }
    int j = lane * 4;
    const float4 xlv = *(const float4*)(xl + (size_t)s * HID + j);
    const float4 xrv = *(const float4*)(xr + (size_t)d * HID + j);
    float ew0 = 0.f, ew1 = 0.f, ew2 = 0.f, ew3 = 0.f;
#pragma unroll
    for (int c = 0; c < EDGE_DIM; c++) {
        float ec = e6[c];
        const float4 wv = *(const float4*)(We + c * HID + j);
        ew0 += ec * wv.x; ew1 += ec * wv.y; ew2 += ec * wv.z; ew3 += ec * wv.w;
    }
    float m0 = lrelu(xlv.x + xrv.x + ew0, 0.2f);
    float m1 = lrelu(xlv.y + xrv.y + ew1, 0.2f);
    float m2 = lrelu(xlv.z + xrv.z + ew2, 0.2f);
    float m3 = lrelu(xlv.w + xrv.w + ew3, 0.2f);
    const float4 av = *(const float4*)(att + j);
    float p = m0 * av.x + m1 * av.y + m2 * av.z + m3 * av.w;
#pragma unroll
    for (int off = 16; off; off >>= 1) p += __shfl_xor(p, off, 32);
    if (lane == 0) {
        score[wid] = p;
        atomicMax(nmax + d, fenc(p));
    }
}

__global__ void k_edge_exp(const int* __restrict__ dst, const float* __restrict__ score,
                           const unsigned* __restrict__ nmax, float* __restrict__ escore,
                           float* __restrict__ denom) {
    int e = blockIdx.x * blockDim.x + threadIdx.x;
    if (e >= EAUG) return;
    int d = (e < N_EDGES) ? dst[e] : (e - N_EDGES);
    float mx = fdec(nmax[d]);
    float v = __expf(score[e] - mx);
    escore[e] = v;
    atomicAdd(denom + d, v);
}

__global__ void k_edge_scatter(const int* __restrict__ src, const int* __restrict__ dst,
                               const float* __restrict__ escore, const float* __restrict__ denom,
                               const float* __restrict__ xl, float* __restrict__ h) {
    int wid  = blockIdx.x * (blockDim.x >> 5) + (threadIdx.x >> 5);
    int lane = threadIdx.x & 31;
    if (wid >= EAUG) return;
    int s, d;
    if (wid < N_EDGES) { s = src[wid]; d = dst[wid]; }
    else               { s = wid - N_EDGES; d = s; }
    float alpha = escore[wid] / (denom[d] + 1e-16f);
    int j = lane * 4;
    const float4 xlv = *(const float4*)(xl + (size_t)s * HID + j);
    float* o = h + (size_t)d * HID + j;
    atomicAdd(o + 0, alpha * xlv.x);
    atomicAdd(o + 1, alpha * xlv.y);
    atomicAdd(o + 2, alpha * xlv.z);
    atomicAdd(o + 3, alpha * xlv.w);
}

// ---------------- GraphNorm + PReLU ----------------
__global__ void k_gn_sum(const float* __restrict__ h, const int* __restrict__ batch,
                         float* gsum) {
    int i = blockIdx.x * blockDim.x + threadIdx.x;
    if (i >= N_NODES * HID) return;
    int node = i >> 7, j = i & (HID - 1);
    atomicAdd(gsum + (size_t)batch[node] * HID + j, h[i]);
}
__global__ void k_gn_center(float* h, const int* __restrict__ batch,
                            const float* __restrict__ gsum, const float* __restrict__ cnt,
                            const float* __restrict__ ms, float* gvar) {
    int i = blockIdx.x * blockDim.x + threadIdx.x;
    if (i >= N_NODES * HID) return;
    int node = i >> 7, j = i & (HID - 1);
    int b = batch[node];
    float c = fmaxf(cnt[b], 1.f);
    float mean = gsum[(size_t)b * HID + j] / c;
    float o = h[i] - mean * ms[j];
    h[i] = o;
    atomicAdd(gvar + (size_t)b * HID + j, o * o);
}
__global__ void k_gn_final(float* h, const int* __restrict__ batch,
                           const float* __restrict__ gvar, const float* __restrict__ cnt,
                           const float* __restrict__ w, const float* __restrict__ bb,
                           const float* __restrict__ pa) {
    int i = blockIdx.x * blockDim.x + threadIdx.x;
    if (i >= N_NODES * HID) return;
    int node = i >> 7, j = i & (HID - 1);
    int b = batch[node];
    float c = fmaxf(cnt[b], 1.f);
    float var = gvar[(size_t)b * HID + j] / c;
    float val = w[j] * h[i] * rsqrtf(var + 1e-5f) + bb[j];
    float a = pa[0];
    h[i] = val >= 0.f ? val : a * val;
}

// ---------------- pooling + head ----------------
__global__ void k_pool_sum(const float* __restrict__ op, const int* __restrict__ batch,
                           float* p1) {
    int i = blockIdx.x * blockDim.x + threadIdx.x;
    if (i >= N_NODES * 64) return;
    int node = i >> 6, j = i & 63;
    atomicAdd(p1 + (size_t)batch[node] * 64 + j, op[i]);
}
__global__ void k_pool_div(float* p1, const float* __restrict__ cnt) {
    int i = blockIdx.x * blockDim.x + threadIdx.x;
    if (i >= NUM_GRAPHS * 64) return;
    p1[i] /= fmaxf(cnt[i >> 6], 1.f);
}
__global__ void k_final(const float* __restrict__ emb, const float* __restrict__ W,
                        const float* __restrict__ b, float* __restrict__ out) {
    int i = blockIdx.x * blockDim.x + threadIdx.x;
    if (i >= NUM_GRAPHS * 3) return;
    int g = i / 3, j = i - g * 3;
    float s = b[j];
#pragma unroll 8
    for (int c = 0; c < 64; c++) s += emb[(size_t)g * 64 + c] * W[c * 3 + j];
    out[i] = s;
}

extern "C" void kernel_launch(void* const* d_in, const int* in_sizes, int n_in,
                              void* d_out, int out_size, void* d_ws, size_t ws_size,
                              hipStream_t stream) {
    // inputs in setup_inputs() order
    const float* x     = (const float*)d_in[0];
    const int*   ei    = (const int*)d_in[1];
    const float* ea    = (const float*)d_in[2];
    const int*   batch = (const int*)d_in[3];
    const float* Wl1 = (const float*)d_in[4];   const float* bl1 = (const float*)d_in[5];
    const float* Wr1 = (const float*)d_in[6];   const float* br1 = (const float*)d_in[7];
    const float* We1 = (const float*)d_in[8];   const float* at1 = (const float*)d_in[9];
    const float* bc1 = (const float*)d_in[10];
    const float* Wl  = (const float*)d_in[11];  const float* bl  = (const float*)d_in[12];
    const float* Wr  = (const float*)d_in[13];  const float* br  = (const float*)d_in[14];
    const float* We  = (const float*)d_in[15];  const float* at  = (const float*)d_in[16];
    const float* bc  = (const float*)d_in[17];
    const float* gnw = (const float*)d_in[18];  const float* gnb = (const float*)d_in[19];
    const float* gnms= (const float*)d_in[20];  const float* pa  = (const float*)d_in[21];
    const float* Wri = (const float*)d_in[22];  const float* bri = (const float*)d_in[23];
    const float* Wro = (const float*)d_in[24];  const float* bro = (const float*)d_in[25];
    const float* Wfc = (const float*)d_in[26];  const float* bfc = (const float*)d_in[27];
    const float* Wfc2= (const float*)d_in[28];  const float* bfc2= (const float*)d_in[29];
    (void)in_sizes; (void)n_in; (void)out_size; (void)ws_size;

    const int* src = ei;
    const int* dst = ei + N_EDGES;

    // workspace carve
    char* wsb = (char*)d_ws;
    size_t off = 0;
    auto carve = [&](size_t bytes) -> void* {
        void* p = wsb + off;
        off += (bytes + 255) & ~(size_t)255;
        return p;
    };
    float*    h      = (float*)carve((size_t)N_NODES * HID * 4);
    float*    xl     = (float*)carve((size_t)N_NODES * HID * 4);
    float*    xr     = (float*)carve((size_t)N_NODES * HID * 4);
    float*    mea    = (float*)carve((size_t)N_NODES * EDGE_DIM * 4);
    float*    deg    = (float*)carve((size_t)N_NODES * 4);
    float*    score  = (float*)carve((size_t)EAUG * 4);
    float*    escore = (float*)carve((size_t)EAUG * 4);
    unsigned* nmax   = (unsigned*)carve((size_t)N_NODES * 4);
    float*    denom  = (float*)carve((size_t)N_NODES * 4);
    float*    gsum   = (float*)carve((size_t)NUM_GRAPHS * HID * 4);
    float*    gvar   = (float*)carve((size_t)NUM_GRAPHS * HID * 4);
    float*    cnt    = (float*)carve((size_t)NUM_GRAPHS * 4);
    float*    p1     = (float*)carve((size_t)NUM_GRAPHS * 64 * 4);
    float*    Wst    = (float*)carve((size_t)HID * HID * 4);   // staged weights (reused, stream-ordered)

    auto gE = [](int n, int b) { return (n + b - 1) / b; };

    auto stage = [&](const float* W, int K, int NO) {
        int total = (NO >> 4) * 32 * (K >> 2);
        k_stage_w<<<gE(total, 256), 256, 0, stream>>>(W, Wst, K, NO);
    };

    // ---- precompute self-loop edge_attr (mean of incoming) and graph counts ----
    k_fill<<<gE(N_NODES, 256), 256, 0, stream>>>(deg, 0.f, N_NODES);
    k_fill<<<gE(N_NODES * EDGE_DIM, 256), 256, 0, stream>>>(mea, 0.f, N_NODES * EDGE_DIM);
    k_fill<<<gE(NUM_GRAPHS, 256), 256, 0, stream>>>(cnt, 0.f, NUM_GRAPHS);
    k_deg_ea<<<gE(N_EDGES, 256), 256, 0, stream>>>(dst, ea, deg, mea);
    k_mea_fin<<<gE(N_NODES * EDGE_DIM, 256), 256, 0, stream>>>(mea, deg);
    k_cnt<<<gE(N_NODES, 256), 256, 0, stream>>>(batch, cnt);

    // ---- 5 GATv2 layers ----
    for (int L = 0; L < 5; L++) {
        const float *Wl_, *bl_, *Wr_, *br_, *We_, *att_, *bc_;
        int K;
        const float* A = (L == 0) ? x : h;
        if (L == 0) {
            Wl_ = Wl1; bl_ = bl1; Wr_ = Wr1; br_ = br1;
            We_ = We1; att_ = at1; bc_ = bc1; K = 48;
        } else {
            int i = L - 1;
            Wl_ = Wl + (size_t)i * HID * HID; bl_ = bl + (size_t)i * HID;
            Wr_ = Wr + (size_t)i * HID * HID; br_ = br + (size_t)i * HID;
            We_ = We + (size_t)i * EDGE_DIM * HID;
            att_ = at + (size_t)i * HID; bc_ = bc + (size_t)i * HID;
            K = HID;
        }
        stage(Wl_, K, HID);
        k_gemm_wmma<8, 0, 0><<<N_NODES / 16, 32, 0, stream>>>(A, Wst, bl_, xl, N_NODES, K);
        stage(Wr_, K, HID);
        k_gemm_wmma<8, 0, 0><<<N_NODES / 16, 32, 0, stream>>>(A, Wst, br_, xr, N_NODES, K);
        // h := bc (accumulator for segment-sum output); softmax state reset
        k_init_bias<<<gE(N_NODES * HID, 256), 256, 0, stream>>>(h, bc_, N_NODES * HID);
        k_fill_u32<<<gE(N_NODES, 256), 256, 0, stream>>>(nmax, 0x007FFFFFu, N_NODES); // enc(-inf)
        k_fill<<<gE(N_NODES, 256), 256, 0, stream>>>(denom, 0.f, N_NODES);
        k_edge_score<<<gE(EAUG, 8), 256, 0, stream>>>(src, dst, ea, mea, xl, xr, We_, att_, score, nmax);
        k_edge_exp<<<gE(EAUG, 256), 256, 0, stream>>>(dst, score, nmax, escore, denom);
        k_edge_scatter<<<gE(EAUG, 8), 256, 0, stream>>>(src, dst, escore, denom, xl, h);
        if (L < 4) {
            k_fill<<<gE(NUM_GRAPHS * HID, 256), 256, 0, stream>>>(gsum, 0.f, NUM_GRAPHS * HID);
            k_fill<<<gE(NUM_GRAPHS * HID, 256), 256, 0, stream>>>(gvar, 0.f, NUM_GRAPHS * HID);
            k_gn_sum<<<gE(N_NODES * HID, 256), 256, 0, stream>>>(h, batch, gsum);
            k_gn_center<<<gE(N_NODES * HID, 256), 256, 0, stream>>>(h, batch, gsum, cnt,
                                                                    gnms + (size_t)L * HID, gvar);
            k_gn_final<<<gE(N_NODES * HID, 256), 256, 0, stream>>>(h, batch, gvar, cnt,
                                                                   gnw + (size_t)L * HID,
                                                                   gnb + (size_t)L * HID, pa);
        }
    }

    // ---- readout: ip = h@Wri+bri ; op = lrelu(ip)@Wro+bro ; mean-pool ; head ----
    stage(Wri, HID, 64);
    k_gemm_wmma<4, 0, 0><<<N_NODES / 16, 32, 0, stream>>>(h, Wst, bri, xl /*ip*/, N_NODES, HID);
    stage(Wro, 64, 64);
    k_gemm_wmma<4, 1, 0><<<N_NODES / 16, 32, 0, stream>>>(xl, Wst, bro, xr /*op*/, N_NODES, 64);
    k_fill<<<gE(NUM_GRAPHS * 64, 256), 256, 0, stream>>>(p1, 0.f, NUM_GRAPHS * 64);
    k_pool_sum<<<gE(N_NODES * 64, 256), 256, 0, stream>>>(xr, batch, p1);
    k_pool_div<<<gE(NUM_GRAPHS * 64, 256), 256, 0, stream>>>(p1, cnt);

    float* out0 = (float*)d_out;                 // [4096, 3]
    float* emb  = out0 + NUM_GRAPHS * 3;         // [4096, 64]
    stage(Wfc, 64, 64);
    k_gemm_wmma<4, 0, 1><<<NUM_GRAPHS / 16, 32, 0, stream>>>(p1, Wst, bfc, emb, NUM_GRAPHS, 64);
    k_final<<<gE(NUM_GRAPHS * 3, 256), 256, 0, stream>>>(emb, Wfc2, bfc2, out0);
}